// CostAdjustedCrossEntropyLoss_20718922236554
// MI455X (gfx1250) — compile-verified
//
#include <hip/hip_runtime.h>
#include <cstdint>
#include <cstddef>

typedef float v4f __attribute__((ext_vector_type(4)));

#if defined(__gfx1250__) && __has_builtin(__builtin_amdgcn_global_load_async_to_lds_b128) && __has_builtin(__builtin_amdgcn_s_wait_asynccnt)
#define USE_ASYNC 1
#else
#define USE_ASYNC 0
#endif

#define WPB     8      // waves per block (256 threads)
#define TILE    128    // floats per wave-tile (32 lanes x float4)
#define DEPTH   8      // async pipeline depth: whole C=1000 row (8 tiles) in flight
#define NBLOCKS 1024   // 8192 waves total -> 8 rows per wave at N=65536

#if USE_ASYNC
// Builtin signature (from hipcc diagnostics): (int4 AS1*, int4 AS3*, imm, imm)
typedef int v4i_ __attribute__((vector_size(4 * sizeof(int))));
typedef __attribute__((address_space(1))) v4i_ as1_v4i;
typedef __attribute__((address_space(3))) v4i_ as3_v4i;

__device__ __forceinline__ void issue_tile(const float* gsrc, float* ldst) {
  // one b128 per lane: wave moves 512B global -> LDS, tracked by ASYNCcnt
  __builtin_amdgcn_global_load_async_to_lds_b128(
      (as1_v4i*)gsrc, (as3_v4i*)ldst, 0, 0);
}
#endif

__device__ __forceinline__ void combine(float& m, float& s, int& ai,
                                        float m2, float s2, int i2) {
  const float ninf = -__builtin_inff();
  float nm = fmaxf(m, m2);
  float sa = (m  > ninf) ? s  * __expf(m  - nm) : 0.0f;
  float sb = (m2 > ninf) ? s2 * __expf(m2 - nm) : 0.0f;
  if (m2 > m || (m2 == m && i2 < ai)) ai = i2;   // first-occurrence tie break
  m = nm;
  s = sa + sb;
}

__global__ __launch_bounds__(WPB * 32)
void cace_rows(const float* __restrict__ logits,
               const float* __restrict__ cw,
               const int*   __restrict__ target,
               const int*   __restrict__ sc,
               float*       __restrict__ partials,
               int N, int C)
{
#if USE_ASYNC
  __shared__ float stage[WPB][DEPTH][TILE];   // 32 KB
#endif
  __shared__ float red[2][WPB];

  const int lane = threadIdx.x & 31;
  const int wid  = threadIdx.x >> 5;
  const int gw   = blockIdx.x * WPB + wid;
  const int nw   = gridDim.x * WPB;
  const int ntiles = (C + TILE - 1) / TILE;
  const int lcol = lane * 4;

  float l2acc = 0.0f, wacc = 0.0f;

  for (int row = gw; row < N; row += nw) {
    const float* rowp = logits + (size_t)row * (size_t)C;
    float m = -__builtin_inff();
    float s = 0.0f;
    int   ai = 0x7fffffff;

#if USE_ASYNC
    // prologue: fill pipeline with DEPTH-1 tiles (7 of the 8 tiles at C=1000)
    #pragma unroll
    for (int p = 0; p < DEPTH - 1; ++p) {
      if (p < ntiles) {
        int c0 = p * TILE + lcol;
        if (c0 < C) issue_tile(rowp + c0, &stage[wid][p][lcol]);
      }
    }
#endif

    for (int t = 0; t < ntiles; ++t) {
      const int col0 = t * TILE + lcol;
#if USE_ASYNC
      const int buf = t & (DEPTH - 1);
      {
        int n = t + DEPTH - 1;
        if (n < ntiles) {
          int c0 = n * TILE + lcol;
          if (c0 < C) issue_tile(rowp + c0, &stage[wid][n & (DEPTH - 1)][lcol]);
        }
        int oa = ntiles - 1 - t;                 // tiles allowed to stay in flight
        if (oa > DEPTH - 1) oa = DEPTH - 1;
        switch (oa) {                            // s_wait_asynccnt needs an immediate
          case 0:  __builtin_amdgcn_s_wait_asynccnt(0); break;
          case 1:  __builtin_amdgcn_s_wait_asynccnt(1); break;
          case 2:  __builtin_amdgcn_s_wait_asynccnt(2); break;
          case 3:  __builtin_amdgcn_s_wait_asynccnt(3); break;
          case 4:  __builtin_amdgcn_s_wait_asynccnt(4); break;
          case 5:  __builtin_amdgcn_s_wait_asynccnt(5); break;
          case 6:  __builtin_amdgcn_s_wait_asynccnt(6); break;
          default: __builtin_amdgcn_s_wait_asynccnt(7); break;
        }
        asm volatile("" ::: "memory");           // keep ds reads below the wait
      }
#endif
      if (col0 < C) {
#if USE_ASYNC
        v4f x = *(const v4f*)&stage[wid][buf][lcol];
#else
        // single-use 262MB stream: don't pollute L2 on the fallback path
        v4f x = __builtin_nontemporal_load((const v4f*)(rowp + col0));
#endif
        if (col0 + 3 < C) {
          #pragma unroll
          for (int k = 0; k < 4; ++k) {
            float xv = x[k];
            if (xv > m) { s = s * __expf(m - xv) + 1.0f; m = xv; ai = col0 + k; }
            else        { s += __expf(xv - m); }
          }
        } else {
          #pragma unroll
          for (int k = 0; k < 4; ++k) {
            if (col0 + k < C) {
              float xv = x[k];
              if (xv > m) { s = s * __expf(m - xv) + 1.0f; m = xv; ai = col0 + k; }
              else        { s += __expf(xv - m); }
            }
          }
        }
      }
    }

    // wave32 butterfly reduction of (max, scaled-sum, argmax)
    #pragma unroll
    for (int off = 16; off > 0; off >>= 1) {
      float m2 = __shfl_xor(m,  off, 32);
      float s2 = __shfl_xor(s,  off, 32);
      int   i2 = __shfl_xor(ai, off, 32);
      combine(m, s, ai, m2, s2, i2);
    }

    if (lane == 0) {
      float lse = m + __logf(s);
      int   tt  = sc[target[row]];                     // super_classes[target[i]]
      float tlp = rowp[tt] - lse;                      // logp[i, tt]
      float w   = cw[(size_t)tt * (size_t)C + ai];     // class_weights[tt, argmax]
      l2acc += -w * tlp;
      wacc  += w;
    }
  }

  if (lane == 0) { red[0][wid] = l2acc; red[1][wid] = wacc; }
  __syncthreads();
  if (threadIdx.x == 0) {
    float a = 0.0f, b = 0.0f;
    #pragma unroll
    for (int i = 0; i < WPB; ++i) { a += red[0][i]; b += red[1][i]; }
    partials[2 * blockIdx.x + 0] = a;
    partials[2 * blockIdx.x + 1] = b;
  }
}

__global__ __launch_bounds__(256)
void cace_final(const float* __restrict__ partials, int nb, float* __restrict__ out)
{
  __shared__ float sl[256], sw[256];
  int t = threadIdx.x;
  float a = 0.0f, b = 0.0f;
  for (int i = t; i < nb; i += 256) { a += partials[2 * i]; b += partials[2 * i + 1]; }
  sl[t] = a; sw[t] = b;
  __syncthreads();
  for (int off = 128; off > 0; off >>= 1) {
    if (t < off) { sl[t] += sl[t + off]; sw[t] += sw[t + off]; }
    __syncthreads();
  }
  if (t == 0) out[0] = sl[0] / sw[0];
}

extern "C" void kernel_launch(void* const* d_in, const int* in_sizes, int n_in,
                              void* d_out, int out_size, void* d_ws, size_t ws_size,
                              hipStream_t stream) {
  (void)n_in; (void)out_size;
  const float* logits = (const float*)d_in[0];   // [N, C] f32
  const float* cwgt   = (const float*)d_in[1];   // [C, C] f32
  const int*   target = (const int*)d_in[2];     // [N]
  const int*   superc = (const int*)d_in[3];     // [C]
  const int N = in_sizes[2];
  const int C = in_sizes[3];

  float* partials = (float*)d_ws;
  int nb = NBLOCKS;
  size_t need = (size_t)nb * 2 * sizeof(float);
  if (need > ws_size) nb = (int)(ws_size / (2 * sizeof(float)));
  if (nb < 1) nb = 1;

  cace_rows<<<nb, WPB * 32, 0, stream>>>(logits, cwgt, target, superc, partials, N, C);
  cace_final<<<1, 256, 0, stream>>>(partials, nb, (float*)d_out);
}